// CrossSparseAggrNet_v2_11931419148616
// MI455X (gfx1250) — compile-verified
//
#include <hip/hip_runtime.h>
#include <hip/hip_bf16.h>
#include <math.h>

// ---------------- problem constants ----------------
#define BB    64     // images
#define NCAP  64     // captions
#define LV    197
#define LT    32
#define CC    512
#define NSP   196
#define NKEEP 98
#define KEEPED 49
#define HID   102
#define LN_EPS 1e-5f

typedef __attribute__((ext_vector_type(2))) float v2f;
typedef __attribute__((ext_vector_type(8))) float v8f;

__device__ __forceinline__ v8f wmma4(v8f acc, v2f a, v2f b) {
    // D = A(16x4,f32) * B(4x16,f32) + C ; exact f32 matrix path on CDNA5
    return __builtin_amdgcn_wmma_f32_16x16x4_f32(false, a, false, b,
                                                 (short)0, acc, false, false);
}

__device__ __forceinline__ float wave_sum(float v) {
    #pragma unroll
    for (int o = 16; o > 0; o >>= 1) v += __shfl_xor(v, o, 32);
    return v;
}
__device__ __forceinline__ float wave_max(float v) {
    #pragma unroll
    for (int o = 16; o > 0; o >>= 1) v = fmaxf(v, __shfl_xor(v, o, 32));
    return v;
}

// ---------------- kernel 1: image token norms + attn_x ----------------
// grid = B*LV blocks, 32 threads (one wave32 per token)
__global__ void k_img_prep(const float* __restrict__ img,
                           float* __restrict__ inv_norm,
                           float* __restrict__ attn_x) {
    const int blk = blockIdx.x;            // b*LV + s
    const int b = blk / LV, s = blk % LV;
    const int lane = threadIdx.x;
    const float* x   = img + (size_t)blk * CC;
    const float* cls = img + (size_t)(b * LV) * CC;
    float ss = 0.f, dc = 0.f, cc = 0.f;
    for (int k = lane; k < CC; k += 32) {
        float xv = x[k], cv = cls[k];
        ss += xv * xv; dc += xv * cv; cc += cv * cv;
    }
    ss = wave_sum(ss); dc = wave_sum(dc); cc = wave_sum(cc);
    if (lane == 0) {
        float inv = 1.f / fmaxf(sqrtf(ss), 1e-12f);
        inv_norm[blk] = inv;
        if (s > 0) {
            float invc = 1.f / fmaxf(sqrtf(cc), 1e-12f);
            attn_x[b * NSP + (s - 1)] = dc * inv * invc;
        }
    }
}

// ---------------- kernel 2: caption normalization ----------------
// grid = NCAP*LT blocks, 32 threads
__global__ void k_cap_norm(const float* __restrict__ cap,
                           float* __restrict__ capn) {
    const int blk = blockIdx.x;            // c*LT + t
    const int lane = threadIdx.x;
    const float* x = cap + (size_t)blk * CC;
    float ss = 0.f;
    for (int k = lane; k < CC; k += 32) { float v = x[k]; ss += v * v; }
    ss = wave_sum(ss);
    float inv = 1.f / fmaxf(sqrtf(ss), 1e-12f);
    float* o = capn + (size_t)blk * CC;
    for (int k = lane; k < CC; k += 32) o[k] = x[k] * inv;
}

// ---------------- kernel 3: per-caption projection  (32x512)@(512x512)+bias ----
// grid = NCAP blocks, 256 threads (8 waves). One wave owns an M-strip of
// 4 N-tiles so the A-fragment is fetched once per k-step and reused by 4 WMMAs.
__global__ void __launch_bounds__(256)
k_capw(const float* __restrict__ capn, const float* __restrict__ W,
       const float* __restrict__ bias, float* __restrict__ out) {
    const int c = blockIdx.x;
    const int wv = threadIdx.x >> 5, lane = threadIdx.x & 31;
    const int half = lane >> 4, r = lane & 15;
    const float* A = capn + (size_t)c * LT * CC;
    float*       O = out  + (size_t)c * LT * CC;
    for (int t = wv; t < 16; t += 8) {       // strip = (mi:2) x (nh:8, 4 N-tiles each)
        const int mi = t & 1, nh = t >> 1;
        const int m = mi * 16 + r;           // < 32
        v8f acc[4] = {};
        for (int k0 = 0; k0 < CC; k0 += 4) {
            const int ka = k0 + 2 * half;
            v2f a;
            a.x = A[(size_t)m * CC + ka];
            a.y = A[(size_t)m * CC + ka + 1];
            #pragma unroll
            for (int q = 0; q < 4; ++q) {
                const int n = (nh * 4 + q) * 16 + r;     // < 512
                v2f bf;
                bf.x = W[(size_t)ka * CC + n];
                bf.y = W[(size_t)(ka + 1) * CC + n];
                acc[q] = wmma4(acc[q], a, bf);
            }
        }
        #pragma unroll
        for (int q = 0; q < 4; ++q) {
            const int n = (nh * 4 + q) * 16 + r;
            #pragma unroll
            for (int v = 0; v < 8; ++v) {
                const int mm = mi * 16 + v + (half ? 8 : 0);
                O[(size_t)mm * CC + n] = acc[q][v] + bias[n];
            }
        }
    }
}

// ---------------- kernel 4: score, rank-based top-98 split, extra vector -----
// grid = NCAP*B blocks (P = c*B+b), 256 threads
__global__ void __launch_bounds__(256)
k_select(const float* __restrict__ img, const float* __restrict__ capn,
         const float* __restrict__ inv_norm, const float* __restrict__ attn_x,
         int* __restrict__ keep_idx, float* __restrict__ extra) {
    const int P = blockIdx.x;
    const int c = P >> 6, b = P & 63;
    __shared__ float score[NSP];
    __shared__ int   nk_idx[NKEEP];
    __shared__ float nk_p[NKEEP];
    const int tid = threadIdx.x, wv = tid >> 5, lane = tid & 31;
    const float* glo = capn + (size_t)c * LT * CC;     // cap_glo = cap_norm[c,0]
    // score = attn_x + <cap_glo, img_sp_norm>
    for (int s = wv; s < NSP; s += 8) {
        const float* x = img + (size_t)(b * LV + 1 + s) * CC;
        float d = 0.f;
        for (int k = lane; k < CC; k += 32) d += x[k] * glo[k];
        d = wave_sum(d);
        if (lane == 0)
            score[s] = attn_x[b * NSP + s] + d * inv_norm[b * LV + 1 + s];
    }
    __syncthreads();
    // rank == position in stable argsort(-score); permutation-invariant downstream
    if (tid < NSP) {
        const float sv = score[tid];
        int rank = 0;
        for (int t = 0; t < NSP; ++t) {
            const float tv = score[t];
            rank += (tv > sv) || (tv == sv && t < tid);
        }
        if (rank < NKEEP) keep_idx[(size_t)P * NKEEP + rank] = tid;
        else { nk_idx[rank - NKEEP] = tid; nk_p[rank - NKEEP] = sv; }
    }
    __syncthreads();
    if (wv == 0) {                                    // softmax over nonkeep scores
        float m = -1e30f;
        for (int j = lane; j < NKEEP; j += 32) m = fmaxf(m, nk_p[j]);
        m = wave_max(m);
        float s = 0.f;
        for (int j = lane; j < NKEEP; j += 32) { float e = expf(nk_p[j] - m); nk_p[j] = e; s += e; }
        s = wave_sum(s);
        const float inv = 1.f / s;
        for (int j = lane; j < NKEEP; j += 32) nk_p[j] *= inv;
    }
    __syncthreads();
    float* ex = extra + (size_t)P * CC;
    for (int ch = tid; ch < CC; ch += 256) {
        float acc = 0.f;
        for (int j = 0; j < NKEEP; ++j)
            acc += nk_p[j] * img[(size_t)(b * LV + 1 + nk_idx[j]) * CC + ch];
        ex[ch] = acc;
    }
}

// ---------------- kernel 5: per-(caption,image) fused pipeline ----------------
// LDS layout (floats); stride 513 on sel_tok kills bank conflicts on B^T reads.
#define ST_STRIDE 513
#define OFF_ST    0                              // 51*513 = 26163  (sel_tok)
#define OFF_H     (OFF_ST + 51 * ST_STRIDE)      // 98*104 = 10192  (h; reused: c2i/z1/z2)
#define OFF_W     (OFF_H + 98 * 104)             // 49*100 = 4900   (softmax wts; reused: stats)
#define OFF_G     (OFF_W + 49 * 100)             // 512 gamma
#define OFF_BT    (OFF_G + CC)                   // 512 beta
#define OFF_MEAN  (OFF_BT + CC)                  // 98
#define OFF_RSTD  (OFF_MEAN + NKEEP)             // 98
#define OFF_KEEP  (OFF_RSTD + NKEEP)             // 98 (int)
#define OFF_RED   (OFF_KEEP + NKEEP)             // 8
#define SMEM_FLOATS (OFF_RED + 8)
#define OFF_CI    OFF_H
#define OFF_Z1    (OFF_H + 32 * 52)
#define OFF_Z2    (OFF_H + 64 * 52)

__global__ void __launch_bounds__(256)
k_main(const float* __restrict__ img, const float* __restrict__ capn,
       const float* __restrict__ capWa, const float* __restrict__ capWb,
       const int* __restrict__ keep_idx, const float* __restrict__ extra,
       const float* __restrict__ gamma, const float* __restrict__ beta,
       const float* __restrict__ w1, const float* __restrict__ b1,
       const float* __restrict__ w2, const float* __restrict__ b2,
       const float* __restrict__ aggr_scale, const float* __restrict__ temp,
       float* __restrict__ out) {
    extern __shared__ float sm[];
    int* keep = (int*)&sm[OFF_KEEP];
    const int P = blockIdx.x;
    const int c = P >> 6, b = P & 63;
    const int tid = threadIdx.x, wv = tid >> 5, lane = tid & 31;
    const int half = lane >> 4, r = lane & 15;
    const float ascale  = aggr_scale[0];
    const float invtemp = 1.f / temp[0];
    const float* imgb = img + (size_t)b * LV * CC;

    if (tid < NKEEP) keep[tid] = keep_idx[(size_t)P * NKEEP + tid];
    for (int k = tid; k < CC; k += 256) { sm[OFF_G + k] = gamma[k]; sm[OFF_BT + k] = beta[k]; }
    __syncthreads();

    // ---- A: layernorm stats for the 98 kept rows ----
    for (int m = wv; m < NKEEP; m += 8) {
        const float* x = imgb + (size_t)(1 + keep[m]) * CC;
        float s = 0.f, ss = 0.f;
        for (int k = lane; k < CC; k += 32) { float v = x[k]; s += v; ss += v * v; }
        s = wave_sum(s); ss = wave_sum(ss);
        if (lane == 0) {
            float mu = s * (1.f / CC);
            sm[OFF_MEAN + m] = mu;
            sm[OFF_RSTD + m] = rsqrtf(ss * (1.f / CC) - mu * mu + LN_EPS);
        }
    }
    __syncthreads();

    // ---- B: h = gelu(ln(sel) @ w1 + b1)  [98x102], K=512.
    // One wave owns an M-strip (mi, 3-4 N-tiles): the gathered+layernormed
    // A-fragment is fetched once per k-step and reused by up to 4 WMMAs.
    for (int t = wv; t < 14; t += 8) {
        const int mi = t % 7, nh = t / 7;          // nh in {0,1}
        const int NT = nh ? 3 : 4;                 // wave-uniform
        const int m = mi * 16 + r;
        const int msafe = (m < NKEEP) ? m : (NKEEP - 1);
        const float* arow = imgb + (size_t)(1 + keep[msafe]) * CC;
        const float mu = sm[OFF_MEAN + msafe], rs = sm[OFF_RSTD + msafe];
        const float mok = (m < NKEEP) ? 1.f : 0.f;
        v8f acc[4] = {};
        for (int k0 = 0; k0 < CC; k0 += 4) {
            const int ka = k0 + 2 * half;
            v2f a;
            a.x = mok * ((arow[ka]     - mu) * rs * sm[OFF_G + ka]     + sm[OFF_BT + ka]);
            a.y = mok * ((arow[ka + 1] - mu) * rs * sm[OFF_G + ka + 1] + sm[OFF_BT + ka + 1]);
            #pragma unroll
            for (int q = 0; q < 4; ++q) {
                if (q < NT) {
                    const int n = (nh * 4 + q) * 16 + r;
                    v2f bf;
                    bf.x = (n < HID) ? w1[(size_t)ka * HID + n]       : 0.f;
                    bf.y = (n < HID) ? w1[(size_t)(ka + 1) * HID + n] : 0.f;
                    acc[q] = wmma4(acc[q], a, bf);
                }
            }
        }
        #pragma unroll
        for (int q = 0; q < 4; ++q) {
            if (q < NT) {
                const int n = (nh * 4 + q) * 16 + r;
                #pragma unroll
                for (int v = 0; v < 8; ++v) {
                    const int mm = mi * 16 + v + (half ? 8 : 0);
                    if (mm < NKEEP && n < HID) {
                        float x = acc[q][v] + b1[n];
                        sm[OFF_H + mm * 104 + n] = 0.5f * x * (1.f + erff(x * 0.70710678118654752f));
                    }
                }
            }
        }
    }
    __syncthreads();

    // ---- C: wls[j,m] = (h @ w2 + b2)[m,j] * aggr_scale   [98x49], K=102->104 ----
    for (int t = wv; t < 28; t += 8) {
        const int mi = t >> 2, ni = t & 3;
        const int m = mi * 16 + r, n = ni * 16 + r;
        v8f acc = {};
        for (int k0 = 0; k0 < 104; k0 += 4) {
            const int ka = k0 + 2 * half;
            v2f a, bf;
            a.x = (m < NKEEP && ka     < HID) ? sm[OFF_H + m * 104 + ka]     : 0.f;
            a.y = (m < NKEEP && ka + 1 < HID) ? sm[OFF_H + m * 104 + ka + 1] : 0.f;
            bf.x = (ka     < HID && n < KEEPED) ? w2[(size_t)ka * KEEPED + n]       : 0.f;
            bf.y = (ka + 1 < HID && n < KEEPED) ? w2[(size_t)(ka + 1) * KEEPED + n] : 0.f;
            acc = wmma4(acc, a, bf);
        }
        #pragma unroll
        for (int v = 0; v < 8; ++v) {
            const int mm = mi * 16 + v + (half ? 8 : 0);
            if (mm < NKEEP && n < KEEPED)
                sm[OFF_W + n * 100 + mm] = (acc[v] + b2[n]) * ascale;
        }
    }
    __syncthreads();

    // softmax over the 98 tokens for each of the 49 aggregation heads
    for (int j = wv; j < KEEPED; j += 8) {
        float mx = -1e30f;
        for (int m2 = lane; m2 < NKEEP; m2 += 32) mx = fmaxf(mx, sm[OFF_W + j * 100 + m2]);
        mx = wave_max(mx);
        float s = 0.f;
        for (int m2 = lane; m2 < NKEEP; m2 += 32) {
            float e = expf(sm[OFF_W + j * 100 + m2] - mx);
            sm[OFF_W + j * 100 + m2] = e; s += e;
        }
        s = wave_sum(s);
        const float inv = 1.f / s;
        for (int m2 = lane; m2 < NKEEP; m2 += 32) sm[OFF_W + j * 100 + m2] *= inv;
    }
    __syncthreads();

    // ---- D: aggr = softw(49x98) @ sel(98x512) -> sel_tok rows 1..49.
    // One wave owns an N-strip: the gathered global B-fragment is fetched once
    // per k-step and reused across all 4 M-tiles (A comes from cheap LDS).
    for (int ni = wv; ni < 32; ni += 8) {
        const int n = ni * 16 + r;
        v8f acc[4] = {};
        for (int k0 = 0; k0 < 100; k0 += 4) {
            const int ka = k0 + 2 * half;
            const int k0s = (ka     < NKEEP) ? ka     : (NKEEP - 1);
            const int k1s = (ka + 1 < NKEEP) ? ka + 1 : (NKEEP - 1);
            v2f bf;
            bf.x = (ka     < NKEEP) ? imgb[(size_t)(1 + keep[k0s]) * CC + n] : 0.f;
            bf.y = (ka + 1 < NKEEP) ? imgb[(size_t)(1 + keep[k1s]) * CC + n] : 0.f;
            #pragma unroll
            for (int mi = 0; mi < 4; ++mi) {
                const int m = mi * 16 + r;
                v2f a;
                a.x = (m < KEEPED && ka     < NKEEP) ? sm[OFF_W + m * 100 + ka]     : 0.f;
                a.y = (m < KEEPED && ka + 1 < NKEEP) ? sm[OFF_W + m * 100 + ka + 1] : 0.f;
                acc[mi] = wmma4(acc[mi], a, bf);
            }
        }
        #pragma unroll
        for (int mi = 0; mi < 4; ++mi) {
            #pragma unroll
            for (int v = 0; v < 8; ++v) {
                const int mm = mi * 16 + v + (half ? 8 : 0);
                if (mm < KEEPED) sm[OFF_ST + (1 + mm) * ST_STRIDE + n] = acc[mi][v];
            }
        }
    }
    __syncthreads();

    // ---- E: rows 0 (cls) and 50 (extra); then row-normalize sel_tok ----
    for (int k = tid; k < CC; k += 256) {
        sm[OFF_ST + 0  * ST_STRIDE + k] = imgb[k];
        sm[OFF_ST + 50 * ST_STRIDE + k] = extra[(size_t)P * CC + k];
    }
    __syncthreads();
    for (int rw = wv; rw < 51; rw += 8) {
        float ss = 0.f;
        for (int k = lane; k < CC; k += 32) { float v = sm[OFF_ST + rw * ST_STRIDE + k]; ss += v * v; }
        ss = wave_sum(ss);
        const float inv = 1.f / fmaxf(sqrtf(ss), 1e-12f);
        for (int k = lane; k < CC; k += 32) sm[OFF_ST + rw * ST_STRIDE + k] *= inv;
    }
    __syncthreads();

    // ---- F: three (32x512)@(512x51) GEMMs against sel_tok^T.
    // One wave owns (mat, mi): the global A-fragment is fetched once per
    // k-step and reused by all 4 N-tiles (B from LDS, stride-513 conflict-free).
    const float* Amats[3] = { capn  + (size_t)c * LT * CC,
                              capWa + (size_t)c * LT * CC,
                              capWb + (size_t)c * LT * CC };
    for (int t = wv; t < 6; t += 8) {          // waves 0..5 active (uniform)
        const int mat = t >> 1, mi = t & 1;
        const float* A = Amats[mat];
        const int m = mi * 16 + r;             // < 32
        v8f acc[4] = {};
        for (int k0 = 0; k0 < CC; k0 += 4) {
            const int ka = k0 + 2 * half;
            v2f a;
            a.x = A[(size_t)m * CC + ka];
            a.y = A[(size_t)m * CC + ka + 1];
            #pragma unroll
            for (int q = 0; q < 4; ++q) {
                const int n = q * 16 + r;
                const int ns = (n < 51) ? n : 50;
                v2f bf;
                bf.x = (n < 51) ? sm[OFF_ST + ns * ST_STRIDE + ka]     : 0.f;
                bf.y = (n < 51) ? sm[OFF_ST + ns * ST_STRIDE + ka + 1] : 0.f;
                acc[q] = wmma4(acc[q], a, bf);
            }
        }
        #pragma unroll
        for (int q = 0; q < 4; ++q) {
            const int n = q * 16 + r;
            if (n < 51) {
                #pragma unroll
                for (int v = 0; v < 8; ++v) {
                    const int mm = mi * 16 + v + (half ? 8 : 0);  // < 32
                    float x = acc[q][v];
                    if (mat == 0) sm[OFF_CI + mm * 52 + n] = (x > 0.f) ? x : 0.1f * x; // leaky_relu
                    else          sm[OFF_H + mat * 32 * 52 + mm * 52 + n] = x * invtemp;
                }
            }
        }
    }
    __syncthreads();

    // ---- G: dual softmaxes + weighted reduction ----
    for (int l = wv; l < 51; l += 8) {          // z1 softmax over 32 rows (axis=1)
        const float v = sm[OFF_Z1 + lane * 52 + l];   // lane = caption token, 32 lanes
        const float mx = wave_max(v);
        const float s  = wave_sum(expf(v - mx));
        if (lane == 0) { sm[OFF_W + l] = mx; sm[OFF_W + 51 + l] = s; }
    }
    for (int tr = wv; tr < 32; tr += 8) {       // z2 softmax over 51 cols (axis=2)
        float mx = -1e30f;
        for (int l = lane; l < 51; l += 32) mx = fmaxf(mx, sm[OFF_Z2 + tr * 52 + l]);
        mx = wave_max(mx);
        float s = 0.f;
        for (int l = lane; l < 51; l += 32) s += expf(sm[OFF_Z2 + tr * 52 + l] - mx);
        s = wave_sum(s);
        if (lane == 0) { sm[OFF_W + 102 + tr] = mx; sm[OFF_W + 134 + tr] = s; }
    }
    __syncthreads();
    float part = 0.f;
    for (int idx = tid; idx < 32 * 51; idx += 256) {
        const int tr = idx / 51, l = idx % 51;
        const float cv = sm[OFF_CI + tr * 52 + l];
        const float a1 = expf(sm[OFF_Z1 + tr * 52 + l] - sm[OFF_W + l])        / sm[OFF_W + 51 + l];
        const float a2 = expf(sm[OFF_Z2 + tr * 52 + l] - sm[OFF_W + 102 + tr]) / sm[OFF_W + 134 + tr];
        part += a1 * cv * (1.f / 51.f) + a2 * cv * (1.f / 32.f);
    }
    part = wave_sum(part);
    if (lane == 0) sm[OFF_RED + wv] = part;
    __syncthreads();
    if (tid == 0) {
        float s = 0.f;
        #pragma unroll
        for (int i = 0; i < 8; ++i) s += sm[OFF_RED + i];
        out[b * NCAP + c] = s;          // sims.T layout: (B, NCAP)
    }
}

// ---------------- host launcher ----------------
extern "C" void kernel_launch(void* const* d_in, const int* in_sizes, int n_in,
                              void* d_out, int out_size, void* d_ws, size_t ws_size,
                              hipStream_t stream) {
    (void)in_sizes; (void)n_in; (void)out_size; (void)ws_size;
    const float* img   = (const float*)d_in[0];
    const float* cap   = (const float*)d_in[1];
    /* d_in[2] = cap_lens (unused; all == LT) */
    const float* gamma = (const float*)d_in[3];
    const float* beta  = (const float*)d_in[4];
    const float* w1    = (const float*)d_in[5];
    const float* b1    = (const float*)d_in[6];
    const float* w2    = (const float*)d_in[7];
    const float* b2    = (const float*)d_in[8];
    const float* ascl  = (const float*)d_in[9];
    const float* wi2t  = (const float*)d_in[10];
    const float* bi2t  = (const float*)d_in[11];
    const float* wt2i  = (const float*)d_in[12];
    const float* bt2i  = (const float*)d_in[13];
    const float* temp  = (const float*)d_in[14];
    float* out = (float*)d_out;

    // workspace carve-up (floats)
    float* ws    = (float*)d_ws;
    float* capn  = ws;                                   // NCAP*LT*CC
    float* capWa = capn  + (size_t)NCAP * LT * CC;       // NCAP*LT*CC
    float* capWb = capWa + (size_t)NCAP * LT * CC;       // NCAP*LT*CC
    float* invn  = capWb + (size_t)NCAP * LT * CC;       // B*LV
    float* attnx = invn  + (size_t)BB * LV;              // B*NSP
    float* extra = attnx + (size_t)BB * NSP;             // NCAP*B*CC
    int*   keep  = (int*)(extra + (size_t)NCAP * BB * CC); // NCAP*B*NKEEP

    const size_t smem_bytes = (size_t)SMEM_FLOATS * sizeof(float); // ~170 KB < 320 KB/WGP
    (void)hipFuncSetAttribute((const void*)k_main,
                              hipFuncAttributeMaxDynamicSharedMemorySize,
                              (int)smem_bytes);

    k_img_prep<<<BB * LV, 32, 0, stream>>>(img, invn, attnx);
    k_cap_norm<<<NCAP * LT, 32, 0, stream>>>(cap, capn);
    k_capw<<<NCAP, 256, 0, stream>>>(capn, wi2t, bi2t, capWa);
    k_capw<<<NCAP, 256, 0, stream>>>(capn, wt2i, bt2i, capWb);
    k_select<<<NCAP * BB, 256, 0, stream>>>(img, capn, invn, attnx, keep, extra);
    k_main<<<NCAP * BB, 256, smem_bytes, stream>>>(img, capn, capWa, capWb,
                                                   keep, extra, gamma, beta,
                                                   w1, b1, w2, b2, ascl, temp, out);
}